// SimpleTAMCaD_31353261260856
// MI455X (gfx1250) — compile-verified
//
#include <hip/hip_runtime.h>

typedef __attribute__((ext_vector_type(2))) float v2f;
typedef __attribute__((ext_vector_type(4))) float v4f;
typedef __attribute__((ext_vector_type(8))) float v8f;
typedef __attribute__((ext_vector_type(4))) unsigned int u32x4;
typedef __attribute__((ext_vector_type(4))) int i32x4;
typedef __attribute__((ext_vector_type(8))) int i32x8;

#define B_   4
#define K_   2
#define NV_  32
#define H_   16
#define T_   4096
#define G_   (K_ * NV_)   // 64
#define OPG_ 96
#define TT_  16           // time tile (WMMA N dimension)

// One workgroup = 8 waves handles one (b,k) pair and a 16-wide time tile.
// Stage 0 : one TDM tensor_load_to_lds DMA pulls the x tile
//           (512 rows x 17 fp32, row stride T) into LDS; tensor_dim0 = T - t0
//           makes the hardware zero-fill the tau==T pad column.
// Phase A : project ctx rows (o in [32,48) and [80,96)) -> LDS ctx[d][t][j]
// Phase B : per i-chunk of 16: project att rows -> LDS att[i][t][j],
//           softmax over j=64 per (i,t), coalesced attn store, then
//           out = att @ ctx via 16-step f32 WMMA chains.
__global__ __launch_bounds__(256) void tamcad_fused_kernel(
    const float* __restrict__ x,     // [B][G*H][T]
    const float* __restrict__ Wm,    // [G][OPG][H]
    const float* __restrict__ bias,  // [G][OPG]
    float* __restrict__ out,         // [B*K][NV][H][T]
    float* __restrict__ attn)        // [B*K][NV][2*NV][T]
{
    __shared__ float lds_x[32][16][17];     // [g][d][s], s=0..16, dense TDM layout
    __shared__ float lds_att[16][16][68];   // [i_local][tl][j]
    __shared__ float lds_ctx[16][16][68];   // [d][tl][j]

    const int tid  = threadIdx.x;
    const int lane = tid & 31;
    const int wave = tid >> 5;
    const int lrow = lane & 15;   // M (A rows) / N (B,D cols) position
    const int lhi  = lane >> 4;   // lanes 16-31 hold K=2,3 of the A/B pairs

    const int t0 = blockIdx.x * TT_;
    const int bk = blockIdx.y;           // 0..7
    const int b  = bk / K_;
    const int k  = bk % K_;

    // ---- warm L2/L0 with this (b,k)'s W slice (32 groups x 96 x 16 fp32 = 192KB) ----
    {
        const float* wbase = Wm + (size_t)k * NV_ * OPG_ * H_;
        for (int p = 0; p < 6; ++p)
            __builtin_prefetch(wbase + (size_t)(p * 256 + tid) * 32, 0, 3);
    }

    // ---- Stage 0: single TDM DMA of the x tile into LDS ----
    // Tile: dim0 = t (17 elements, contiguous), dim1 = 512 rows (stride T).
    // tensor_dim0 = T - t0 => the 17th column of the last tile is OOB and
    // returns zero (the reference's pad column).
    {
        const unsigned lds_x_addr = (unsigned)(size_t)(&lds_x[0][0][0]);
        const float* tile_base =
            x + ((size_t)b * (G_ * H_) + (size_t)k * (NV_ * H_)) * (size_t)T_ + t0;
        const unsigned long long ga = (unsigned long long)(size_t)tile_base;
        const unsigned dim0 = (unsigned)(T_ - t0);   // remaining valid t from tile start

        u32x4 g0;
        g0.x = 1u;                                   // count = 1 (valid descriptor)
        g0.y = lds_x_addr;                           // lds_addr
        g0.z = (unsigned)ga;                         // global_addr[31:0]
        g0.w = (unsigned)((ga >> 32) & 0x01FFFFFFu)  // global_addr[56:32]
             | 0x80000000u;                          // type = 2 ("image") in bits 127:126

        i32x8 g1;
        g1[0] = 0x00020000;                          // data_size = 2 (4 bytes)
        g1[1] = (int)(dim0 << 16);                   // tensor_dim0[15:0] -> bits 63:48
        g1[2] = (int)(((dim0 >> 16) & 0xFFFFu)      // tensor_dim0[31:16] -> bits 79:64
             | (512u << 16));                        // tensor_dim1[15:0] -> bits 95:80
        g1[3] = (int)(17u << 16);                    // tile_dim0 = 17 -> bits 127:112
        g1[4] = 512;                                 // tile_dim1 = 512 -> bits 143:128
        g1[5] = T_;                                  // tensor_dim0_stride = 4096
        g1[6] = 0;
        g1[7] = 0;

        i32x4 z4 = {0, 0, 0, 0};
#if __clang_major__ >= 23
        i32x8 z8 = {0, 0, 0, 0, 0, 0, 0, 0};
#endif
        if (wave == 0) {
#if __has_builtin(__builtin_amdgcn_tensor_load_to_lds)
# if __clang_major__ >= 23
            __builtin_amdgcn_tensor_load_to_lds(g0, g1, z4, z4, z8, 0);
# else
            __builtin_amdgcn_tensor_load_to_lds(g0, g1, z4, z4, 0);
# endif
#else
            asm volatile("tensor_load_to_lds %0, %1, %2, %3"
                         :: "s"(g0), "s"(g1), "s"(z4), "s"(z4) : "memory");
#endif
#if __has_builtin(__builtin_amdgcn_s_wait_tensorcnt)
            __builtin_amdgcn_s_wait_tensorcnt(0);
#else
            asm volatile("s_wait_tensorcnt 0x0" ::: "memory");
#endif
        }
    }
    __syncthreads();

    // ---------------- Phase A: ctx projection ----------------
    // 64 tile jobs: (v, half). o0 = 32 (half 0) or 80 (half 1). 8 jobs per wave.
    for (int n = 0; n < 8; ++n) {
        const int job  = wave * 8 + n;      // 0..63
        const int v    = job >> 1;
        const int half = job & 1;
        const int g    = k * NV_ + v;
        const int o0   = 32 + half * 48;    // 32 or 80
        const int s    = half + lrow;       // B-operand time column in staged tile
        v8f acc = {};
        for (int kk = 0; kk < 4; ++kk) {
            const int d0 = kk * 4 + lhi * 2;
            const float* wp = Wm + ((size_t)g * OPG_ + (o0 + lrow)) * H_ + d0;
            v2f a;  a.x  = wp[0];             a.y  = wp[1];
            v2f bb; bb.x = lds_x[v][d0][s];   bb.y = lds_x[v][d0 + 1][s];
            acc = __builtin_amdgcn_wmma_f32_16x16x4_f32(
                false, a, false, bb, (short)0, acc, false, false);
        }
        // D[r] -> o = o0 + r + 8*lhi ; ctx[d][tl][j], j = half*32 + v
        const int j = half * 32 + v;
        for (int r = 0; r < 8; ++r) {
            const int d = r + 8 * lhi;
            lds_ctx[d][lrow][j] = acc[r] + bias[g * OPG_ + (o0 + d)];
        }
    }
    __syncthreads();

    // ---------------- Phase B: two i-chunks of 16 rows ----------------
    for (int ic = 0; ic < 2; ++ic) {
        // --- att projection: 64 jobs = vloc(16) x half(2) x otile(2) ---
        for (int n = 0; n < 8; ++n) {
            const int job  = wave * 8 + n;          // 0..63
            const int vloc = job >> 2;              // 0..15
            const int sub  = job & 3;
            const int half = sub >> 1;
            const int o0   = half * 48 + (sub & 1) * 16;  // 0,16,48,64
            const int v    = ic * 16 + vloc;
            const int g    = k * NV_ + v;
            const int s    = half + lrow;
            v8f acc = {};
            for (int kk = 0; kk < 4; ++kk) {
                const int d0 = kk * 4 + lhi * 2;
                const float* wp = Wm + ((size_t)g * OPG_ + (o0 + lrow)) * H_ + d0;
                v2f a;  a.x  = wp[0];             a.y  = wp[1];
                v2f bb; bb.x = lds_x[v][d0][s];   bb.y = lds_x[v][d0 + 1][s];
                acc = __builtin_amdgcn_wmma_f32_16x16x4_f32(
                    false, a, false, bb, (short)0, acc, false, false);
            }
            // o = o0 + r + 8*lhi ; att[i_local][tl][j], j = half*32 + (o - half*48)
            for (int r = 0; r < 8; ++r) {
                const int o = o0 + r + 8 * lhi;
                const int j = half * 32 + (o - half * 48);
                lds_att[vloc][lrow][j] = acc[r] + bias[g * OPG_ + o];
            }
        }
        __syncthreads();

        // --- softmax over contiguous j=64 row, one (i_local, tl) per thread ---
        {
            const int iloc = tid >> 4;   // 0..15
            const int tl   = tid & 15;
            float* rowp = &lds_att[iloc][tl][0];
            float m = -3.0e38f;
            for (int j = 0; j < 64; ++j) m = fmaxf(m, rowp[j]);
            float ssum = 0.0f;
            for (int j = 0; j < 64; ++j) {
                const float e = __expf(rowp[j] - m);   // TEMPERATURE == 1
                rowp[j] = e;
                ssum += e;
            }
            const float inv = 1.0f / ssum;
            for (int j = 0; j < 64; ++j) rowp[j] *= inv;
        }
        __syncthreads();

        // --- coalesced attn store: 1024 (i,j) rows of 16 t values each ---
        for (int n = 0; n < 4; ++n) {
            const int row  = tid + n * 256;     // 0..1023
            const int iloc = row >> 6;
            const int j    = row & 63;
            const int i    = ic * 16 + iloc;
            float* ap = attn + (((size_t)bk * NV_ + i) * 64 + j) * T_ + t0;
            for (int q = 0; q < 16; q += 4) {
                v4f vv;
                vv.x = lds_att[iloc][q + 0][j];
                vv.y = lds_att[iloc][q + 1][j];
                vv.z = lds_att[iloc][q + 2][j];
                vv.w = lds_att[iloc][q + 3][j];
                *(v4f*)(ap + q) = vv;
            }
        }

        // --- out = att(16x64) @ ctx(64x16), one GEMM per time step ---
        for (int n = 0; n < 2; ++n) {
            const int tl = wave * 2 + n;        // 0..15
            v8f acc = {};
            for (int kk = 0; kk < 16; ++kk) {
                const int j0 = kk * 4 + lhi * 2;
                v2f a;  a.x  = lds_att[lrow][tl][j0];  a.y  = lds_att[lrow][tl][j0 + 1];
                v2f bb; bb.x = lds_ctx[lrow][tl][j0];  bb.y = lds_ctx[lrow][tl][j0 + 1];
                acc = __builtin_amdgcn_wmma_f32_16x16x4_f32(
                    false, a, false, bb, (short)0, acc, false, false);
            }
            // D[r]: i = ic*16 + r + 8*lhi, d = lrow
            const int d = lrow;
            for (int r = 0; r < 8; ++r) {
                const int i = ic * 16 + r + 8 * lhi;
                out[(((size_t)bk * NV_ + i) * H_ + d) * T_ + t0 + tl] = acc[r];
            }
        }
        __syncthreads();   // protect lds_att before next i-chunk overwrites it
    }
}

extern "C" void kernel_launch(void* const* d_in, const int* in_sizes, int n_in,
                              void* d_out, int out_size, void* d_ws, size_t ws_size,
                              hipStream_t stream) {
    const float* x    = (const float*)d_in[0];   // (B, G*H, T)
    const float* Wm   = (const float*)d_in[1];   // (G, OPG, H)
    const float* bias = (const float*)d_in[2];   // (G, OPG)

    float* out  = (float*)d_out;                             // B*K*NV*H*T floats
    float* attn = out + (size_t)B_ * K_ * NV_ * H_ * T_;     // B*K*NV*2NV*T floats

    dim3 grid(T_ / TT_, B_ * K_);   // (256, 8)
    dim3 block(256);
    tamcad_fused_kernel<<<grid, block, 0, stream>>>(x, Wm, bias, out, attn);
}